// AlternateWeaveLayer_14602888806815
// MI455X (gfx1250) — compile-verified
//
#include <hip/hip_runtime.h>

// CDNA5 / gfx1250, wave32. Fp32 WMMA (V_WMMA_F32_16X16X4_F32) GEMM tiles,
// weights staged in LDS, BatchNorm via two-pass recompute (cheaper than
// spilling 256MB of activations to scratch at 23.3 TB/s HBM).

typedef __attribute__((ext_vector_type(2))) float v2f;
typedef __attribute__((ext_vector_type(8))) float v8f;

__device__ __forceinline__ v8f wmma4(v2f a, v2f b, v8f c) {
  // 8-arg form: (neg_a, A, neg_b, B, c_mod, C, reuse_a, reuse_b)
  return __builtin_amdgcn_wmma_f32_16x16x4_f32(
      false, a, false, b, (short)0, c, false, false);
}

__global__ void zero_kernel(float* __restrict__ p, int n) {
  int i = blockIdx.x * blockDim.x + threadIdx.x;
  if (i < n) p[i] = 0.0f;
}

// ---------------------------------------------------------------------------
// Pass 1: h = ReLU(A[rows,64] @ W[64,64] + b); accumulate per-feature
// sum / sumsq (biased batch stats). One wave = one 16x64 output tile.
// ---------------------------------------------------------------------------
__global__ __launch_bounds__(256) void gemm_relu_stats_kernel(
    const float* __restrict__ A, const float* __restrict__ W,
    const float* __restrict__ bias,
    float* __restrict__ gsum, float* __restrict__ gsq, int ntiles) {
  __shared__ v2f   wl[32 * 64];   // W as (k-pair, col) float2 : 16KB
  __shared__ float ssum[64];
  __shared__ float ssq[64];
  const int tid = threadIdx.x;

  for (int i = tid; i < 32 * 64; i += 256) {
    int p = i >> 6, c = i & 63;
    v2f w;
    w.x = W[(2 * p) * 64 + c];
    w.y = W[(2 * p + 1) * 64 + c];
    wl[i] = w;
  }
  if (tid < 64) { ssum[tid] = 0.0f; ssq[tid] = 0.0f; }
  __syncthreads();

  const int lane = tid & 31;
  const int wv   = tid >> 5;
  const int tile = blockIdx.x * 8 + wv;   // wave-uniform
  if (tile < ntiles) {
    const int lo = lane & 15;
    const int hi = lane >> 4;             // 0: K{0,1}, 1: K{2,3}
    const float* arow = A + (size_t)(tile * 16 + lo) * 64;

    v8f acc[4];
#pragma unroll
    for (int nt = 0; nt < 4; ++nt) {
      float bv = bias[nt * 16 + lo];
#pragma unroll
      for (int i = 0; i < 8; ++i) acc[nt][i] = bv;
    }
#pragma unroll
    for (int kk = 0; kk < 16; ++kk) {
      const int k0 = kk * 4;
      v2f a = *(const v2f*)(arow + k0 + hi * 2);
      const int pidx = (k0 >> 1) + hi;
#pragma unroll
      for (int nt = 0; nt < 4; ++nt)
        acc[nt] = wmma4(a, wl[pidx * 64 + nt * 16 + lo], acc[nt]);
    }
    // each lane's 8 accumulator values share one output column
#pragma unroll
    for (int nt = 0; nt < 4; ++nt) {
      float s = 0.0f, q = 0.0f;
#pragma unroll
      for (int i = 0; i < 8; ++i) {
        float v = fmaxf(acc[nt][i], 0.0f);
        s += v; q += v * v;
      }
      const int col = nt * 16 + lo;
      atomicAdd(&ssum[col], s);   // ds_add_f32
      atomicAdd(&ssq[col], q);
    }
  }
  __syncthreads();
  if (tid < 64) {
    atomicAdd(&gsum[tid], ssum[tid]);
    atomicAdd(&gsq[tid], ssq[tid]);
  }
}

// ---------------------------------------------------------------------------
// Fold stats into per-feature affine: scale = g*rsqrt(var+eps),
// shift = be - mean*scale.   sums: [sumA|sqA|sumP|sqP], coef likewise.
// ---------------------------------------------------------------------------
__global__ void stats_finalize_kernel(
    const float* __restrict__ sums, float* __restrict__ coef,
    const float* __restrict__ g_atom, const float* __restrict__ be_atom,
    const float* __restrict__ g_pair, const float* __restrict__ be_pair,
    float invNA, float invNP) {
  const int c = threadIdx.x;
  if (c < 64) {
    float mA = sums[c] * invNA;
    float vA = fmaxf(sums[64 + c] * invNA - mA * mA, 0.0f);
    float sA = g_atom[c] * rsqrtf(vA + 1e-5f);
    coef[c]      = sA;
    coef[64 + c] = be_atom[c] - mA * sA;

    float mP = sums[128 + c] * invNP;
    float vP = fmaxf(sums[192 + c] * invNP - mP * mP, 0.0f);
    float sP = g_pair[c] * rsqrtf(vP + 1e-5f);
    coef[128 + c] = sP;
    coef[192 + c] = be_pair[c] - mP * sP;
  }
}

// ---------------------------------------------------------------------------
// Atom output: recompute GEMM, out = ReLU(xW+b)*scale + shift
// ---------------------------------------------------------------------------
__global__ __launch_bounds__(256) void gemm_relu_bn_out_kernel(
    const float* __restrict__ A, const float* __restrict__ W,
    const float* __restrict__ bias, const float* __restrict__ scale,
    const float* __restrict__ shift, float* __restrict__ out, int ntiles) {
  __shared__ v2f wl[32 * 64];
  const int tid = threadIdx.x;
  for (int i = tid; i < 32 * 64; i += 256) {
    int p = i >> 6, c = i & 63;
    v2f w;
    w.x = W[(2 * p) * 64 + c];
    w.y = W[(2 * p + 1) * 64 + c];
    wl[i] = w;
  }
  __syncthreads();

  const int lane = tid & 31;
  const int tile = blockIdx.x * 8 + (tid >> 5);
  if (tile >= ntiles) return;
  const int lo = lane & 15;
  const int hi = lane >> 4;
  const float* arow = A + (size_t)(tile * 16 + lo) * 64;

  v8f acc[4];
#pragma unroll
  for (int nt = 0; nt < 4; ++nt) {
    float bv = bias[nt * 16 + lo];
#pragma unroll
    for (int i = 0; i < 8; ++i) acc[nt][i] = bv;
  }
#pragma unroll
  for (int kk = 0; kk < 16; ++kk) {
    const int k0 = kk * 4;
    v2f a = *(const v2f*)(arow + k0 + hi * 2);
    const int pidx = (k0 >> 1) + hi;
#pragma unroll
    for (int nt = 0; nt < 4; ++nt)
      acc[nt] = wmma4(a, wl[pidx * 64 + nt * 16 + lo], acc[nt]);
  }
#pragma unroll
  for (int nt = 0; nt < 4; ++nt) {
    const int col = nt * 16 + lo;
    const float sc = scale[col], sh = shift[col];
    const int rbase = tile * 16 + hi * 8;   // C layout: vgpr i -> M=i / M=8+i
#pragma unroll
    for (int i = 0; i < 8; ++i) {
      float v = fmaxf(acc[nt][i], 0.0f);
      out[(size_t)(rbase + i) * 64 + col] = v * sc + sh;
    }
  }
}

// ---------------------------------------------------------------------------
// Pair output: per 16-edge tile gather send/recv rows of x (L2-resident),
// a2p = [send|recv] @ W_a2p (K=128) and pair_bn recompute (K=64), fused.
// 12 v_wmma per k-step, 192 per tile.
// ---------------------------------------------------------------------------
__global__ __launch_bounds__(256) void pair_out_kernel(
    const float* __restrict__ x, const float* __restrict__ pf,
    const int* __restrict__ pairIdx,
    const float* __restrict__ Wp,   const float* __restrict__ bp,
    const float* __restrict__ Wa2p, const float* __restrict__ ba2p,
    const float* __restrict__ scaleP, const float* __restrict__ shiftP,
    float* __restrict__ out2, int E, int ntiles) {
  __shared__ v2f wa[64 * 64];   // W_a2p (128x64) as 64 k-pairs : 32KB
  __shared__ v2f wp[32 * 64];   // W_pair (64x64) as 32 k-pairs : 16KB
  const int tid = threadIdx.x;
  for (int i = tid; i < 64 * 64; i += 256) {
    int p = i >> 6, c = i & 63;
    v2f w;
    w.x = Wa2p[(2 * p) * 64 + c];
    w.y = Wa2p[(2 * p + 1) * 64 + c];
    wa[i] = w;
  }
  for (int i = tid; i < 32 * 64; i += 256) {
    int p = i >> 6, c = i & 63;
    v2f w;
    w.x = Wp[(2 * p) * 64 + c];
    w.y = Wp[(2 * p + 1) * 64 + c];
    wp[i] = w;
  }
  __syncthreads();

  const int lane = tid & 31;
  const int tile = blockIdx.x * 8 + (tid >> 5);
  if (tile >= ntiles) return;
  const int lo = lane & 15;
  const int hi = lane >> 4;
  const int edge = tile * 16 + lo;
  const float* rowS = x + (size_t)pairIdx[edge] * 64;       // send gather
  const float* rowR = x + (size_t)pairIdx[E + edge] * 64;   // recv gather
  const float* rowP = pf + (size_t)edge * 64;

  v8f accA[4], accP[4];
#pragma unroll
  for (int nt = 0; nt < 4; ++nt) {
    float ba = ba2p[nt * 16 + lo];
    float bb = bp[nt * 16 + lo];
#pragma unroll
    for (int i = 0; i < 8; ++i) { accA[nt][i] = ba; accP[nt][i] = bb; }
  }
#pragma unroll
  for (int kk = 0; kk < 16; ++kk) {
    const int k0 = kk * 4;
    v2f aS = *(const v2f*)(rowS + k0 + hi * 2);
    v2f aR = *(const v2f*)(rowR + k0 + hi * 2);
    v2f aP = *(const v2f*)(rowP + k0 + hi * 2);
    const int pidx = (k0 >> 1) + hi;
#pragma unroll
    for (int nt = 0; nt < 4; ++nt) {
      const int bo = nt * 16 + lo;
      accA[nt] = wmma4(aS, wa[pidx * 64 + bo], accA[nt]);          // K 0..63
      accA[nt] = wmma4(aR, wa[(32 + pidx) * 64 + bo], accA[nt]);   // K 64..127
      accP[nt] = wmma4(aP, wp[pidx * 64 + bo], accP[nt]);
    }
  }
#pragma unroll
  for (int nt = 0; nt < 4; ++nt) {
    const int col = nt * 16 + lo;
    const float sc = scaleP[col], sh = shiftP[col];
    const int rbase = tile * 16 + hi * 8;
#pragma unroll
    for (int i = 0; i < 8; ++i) {
      float h = fmaxf(accP[nt][i], 0.0f);
      out2[(size_t)(rbase + i) * 64 + col] = accA[nt][i] + h * sc + sh;
    }
  }
}

// ---------------------------------------------------------------------------
extern "C" void kernel_launch(void* const* d_in, const int* in_sizes, int n_in,
                              void* d_out, int out_size, void* d_ws,
                              size_t ws_size, hipStream_t stream) {
  const float* x       = (const float*)d_in[0];
  const float* pf      = (const float*)d_in[1];
  const int*   pidx    = (const int*)d_in[2];
  const float* W_atom  = (const float*)d_in[3];
  const float* b_atom  = (const float*)d_in[4];
  const float* g_atom  = (const float*)d_in[5];
  const float* be_atom = (const float*)d_in[6];
  const float* W_pair  = (const float*)d_in[7];
  const float* b_pair  = (const float*)d_in[8];
  const float* g_pair  = (const float*)d_in[9];
  const float* be_pair = (const float*)d_in[10];
  const float* W_a2p   = (const float*)d_in[11];
  const float* b_a2p   = (const float*)d_in[12];

  const int N = in_sizes[0] / 64;   // 100000
  const int E = in_sizes[1] / 64;   // 1000000

  float* sums = (float*)d_ws;       // 256 floats: sumA|sqA|sumP|sqP
  float* coef = sums + 256;         // 256 floats: scaleA|shiftA|scaleP|shiftP
  float* out1 = (float*)d_out;
  float* out2 = out1 + (size_t)N * 64;

  const int tilesA = N / 16;
  const int tilesP = E / 16;

  zero_kernel<<<1, 256, 0, stream>>>(sums, 256);
  gemm_relu_stats_kernel<<<(tilesA + 7) / 8, 256, 0, stream>>>(
      x, W_atom, b_atom, sums, sums + 64, tilesA);
  gemm_relu_stats_kernel<<<(tilesP + 7) / 8, 256, 0, stream>>>(
      pf, W_pair, b_pair, sums + 128, sums + 192, tilesP);
  stats_finalize_kernel<<<1, 64, 0, stream>>>(
      sums, coef, g_atom, be_atom, g_pair, be_pair, 1.0f / N, 1.0f / E);
  gemm_relu_bn_out_kernel<<<(tilesA + 7) / 8, 256, 0, stream>>>(
      x, W_atom, b_atom, coef, coef + 64, out1, tilesA);
  pair_out_kernel<<<(tilesP + 7) / 8, 256, 0, stream>>>(
      x, pf, pidx, W_pair, b_pair, W_a2p, b_a2p, coef + 128, coef + 192,
      out2, E, tilesP);
}